// Transformer_12970801234650
// MI455X (gfx1250) — compile-verified
//
#include <hip/hip_runtime.h>
#include <hip/hip_bf16.h>
#include <math.h>

// MI455X / gfx1250, wave32. fp32 WMMA path (V_WMMA_F32_16X16X4_F32).
// Model: BS=256, M=14, D=128, NL=2, H=4, DFF=512; seq-len==1 so attention
// collapses exactly to V->O projections (Q/K unused).

typedef __attribute__((ext_vector_type(2))) float v2f;
typedef __attribute__((ext_vector_type(8))) float v8f;

#define D_  128
#define DFF_ 512
#define MLEN 14

// One 16x16 fp32 C-tile of  Y = X(16 x kdim) * W^T,  W row-major (N x kdim).
// A frag (16x4 f32): lane<16 -> M=lane, K={k0,k0+1}; lane>=16 -> M=lane-16, K={k0+2,k0+3}
// B frag (4x16 f32): same K-pair split across lane halves, N = n0 + (lane&15)
__device__ __forceinline__ v8f gemm_tile(const float* __restrict__ Xlds, int ldx,
                                         const float* __restrict__ W, int ldw,
                                         int n0, int kdim, int lane) {
    const int m  = lane & 15;
    const int hi = lane >> 4;              // 0 or 1
    const float* ar = Xlds + m * ldx + 2 * hi;
    const float* br = W + (size_t)(n0 + m) * ldw + 2 * hi;
    v8f c = {};
    for (int k0 = 0; k0 < kdim; k0 += 4) {
        v2f a = *(const v2f*)(ar + k0);    // ds_load_b64
        v2f b = *(const v2f*)(br + k0);    // global_load_b64 (L2-resident)
        c = __builtin_amdgcn_wmma_f32_16x16x4_f32(false, a, false, b,
                                                  (short)0, c, false, false);
    }
    return c;
}

// Row-wise norm, ddof=1, eps added to std (matches reference _norm).
__device__ __forceinline__ void norm16x128(const float* __restrict__ src_lds,
                                           float* __restrict__ dst_lds,
                                           const float* __restrict__ ga,
                                           const float* __restrict__ gb,
                                           float* red, int tid) {
    if (tid < 16) {
        const float* row = src_lds + tid * D_;
        float mu = 0.f;
        for (int d = 0; d < D_; ++d) mu += row[d];
        mu *= (1.f / 128.f);
        float var = 0.f;
        for (int d = 0; d < D_; ++d) { float z = row[d] - mu; var += z * z; }
        var *= (1.f / 127.f);              // ddof = 1
        red[tid * 2 + 0] = mu;
        red[tid * 2 + 1] = 1.f / (sqrtf(var) + 1e-6f);
    }
    __syncthreads();
    for (int idx = tid; idx < 16 * D_; idx += 256) {
        int m = idx >> 7, d = idx & 127;
        dst_lds[idx] = ga[d] * (src_lds[idx] - red[m * 2]) * red[m * 2 + 1] + gb[d];
    }
    __syncthreads();
}

__global__ __launch_bounds__(256) void transformer_fused(
    const float* __restrict__ src,  const int* __restrict__ tptr,
    const float* __restrict__ fc_w, const float* __restrict__ fc_b,
    const float* __restrict__ gc_w, const float* __restrict__ gc_b,
    const float* __restrict__ ln_w, const float* __restrict__ ln_b,
    const float* __restrict__ vw,   const float* __restrict__ vb,
    const float* __restrict__ ow,   const float* __restrict__ ob,
    const float* __restrict__ n1a,  const float* __restrict__ n1b,
    const float* __restrict__ n2a,  const float* __restrict__ n2b,
    const float* __restrict__ w1,   const float* __restrict__ b1,
    const float* __restrict__ w2,   const float* __restrict__ b2,
    const float* __restrict__ fna,  const float* __restrict__ fnb,
    const float* __restrict__ out_w,const float* __restrict__ out_b,
    float* __restrict__ out) {
    __shared__ __align__(16) float s_src[16 * MLEN];
    __shared__ __align__(16) float s_x [16 * D_];
    __shared__ __align__(16) float s_x2[16 * D_];
    __shared__ __align__(16) float s_t [16 * D_];
    __shared__ __align__(16) float s_h [16 * DFF_];
    __shared__ float s_red[32];

    const int tid  = threadIdx.x;
    const int lane = tid & 31;
    const int wv   = tid >> 5;             // 0..7
    const int hi   = lane >> 4;
    const int r0   = blockIdx.x * 16;      // 16 rows per block, 16 blocks

    // ---- stage 0: load src tile ----
    for (int i = tid; i < 16 * MLEN; i += 256)
        s_src[i] = src[(size_t)(r0 + i / MLEN) * MLEN + (i % MLEN)];
    __syncthreads();

    // ---- conv1d(feat) * sigmoid(conv1d(gate)), mean over length ----
    for (int idx = tid; idx < 16 * D_; idx += 256) {
        int m = idx >> 7, d = idx & 127;
        float fw0 = fc_w[d * 3 + 0], fw1 = fc_w[d * 3 + 1], fw2 = fc_w[d * 3 + 2];
        float gw0 = gc_w[d * 3 + 0], gw1 = gc_w[d * 3 + 1], gw2 = gc_w[d * 3 + 2];
        float fb = fc_b[d], gb_ = gc_b[d];
        const float* srow = s_src + m * MLEN;
        float acc = 0.f;
        for (int t = 0; t < MLEN; ++t) {
            float sm1 = (t > 0)        ? srow[t - 1] : 0.f;
            float s0  = srow[t];
            float sp1 = (t < MLEN - 1) ? srow[t + 1] : 0.f;
            float f = fw0 * sm1 + fw1 * s0 + fw2 * sp1 + fb;
            float g = gw0 * sm1 + gw1 * s0 + gw2 * sp1 + gb_;
            acc += f * (1.f / (1.f + expf(-g)));
        }
        s_x[idx] = acc * (1.f / (float)MLEN);
    }
    __syncthreads();

    // ---- input LayerNorm (ddof=0, eps inside sqrt) + positional encoding ----
    if (tid < 16) {
        const float* row = s_x + tid * D_;
        float mu = 0.f;
        for (int d = 0; d < D_; ++d) mu += row[d];
        mu *= (1.f / 128.f);
        float var = 0.f;
        for (int d = 0; d < D_; ++d) { float z = row[d] - mu; var += z * z; }
        var *= (1.f / 128.f);
        s_red[tid * 2 + 0] = mu;
        s_red[tid * 2 + 1] = 1.f / sqrtf(var + 1e-5f);
    }
    __syncthreads();
    {
        const float tval = (float)tptr[0];
        const float sq   = sqrtf(128.f);
        for (int idx = tid; idx < 16 * D_; idx += 256) {
            int m = idx >> 7, d = idx & 127;
            float xn  = ln_w[d] * (s_x[idx] - s_red[m * 2]) * s_red[m * 2 + 1] + ln_b[d];
            float dev = (float)(d & ~1);   // even index for this (sin,cos) pair
            float ang = tval * expf(dev * (-logf(10000.f) / 128.f));
            float pe  = (d & 1) ? cosf(ang) : sinf(ang);
            s_x[idx]  = xn * sq + pe;
        }
    }
    __syncthreads();

    // ---- transformer layers (attention == V->O projection, seq len 1) ----
    for (int L = 0; L < 2; ++L) {
        const float* vwL = vw + (size_t)L * D_ * D_;   const float* vbL = vb + L * D_;
        const float* owL = ow + (size_t)L * D_ * D_;   const float* obL = ob + L * D_;
        const float* w1L = w1 + (size_t)L * DFF_ * D_; const float* b1L = b1 + L * DFF_;
        const float* w2L = w2 + (size_t)L * D_ * DFF_; const float* b2L = b2 + L * D_;

        norm16x128(s_x, s_x2, n1a + L * D_, n1b + L * D_, s_red, tid);

        // warm L2/L0 for the FFN weights while attention GEMMs run
        for (int off = tid; off < (DFF_ * D_) / 64; off += 256) {
            __builtin_prefetch(w1L + off * 64, 0, 1);
            __builtin_prefetch(w2L + off * 64, 0, 1);
        }

        // v = x2 @ vw^T + vb  -> s_t        (16x128x128; 8 waves x 1 N-tile)
        {
            v8f c = gemm_tile(s_x2, D_, vwL, D_, wv * 16, D_, lane);
            int n = wv * 16 + (lane & 15);
            float bias = vbL[n];
            #pragma unroll
            for (int j = 0; j < 8; ++j) s_t[(j + 8 * hi) * D_ + n] = c[j] + bias;
        }
        __syncthreads();

        // x += v @ ow^T + ob               (residual folded into C store)
        {
            v8f c = gemm_tile(s_t, D_, owL, D_, wv * 16, D_, lane);
            int n = wv * 16 + (lane & 15);
            float bias = obL[n];
            #pragma unroll
            for (int j = 0; j < 8; ++j) s_x[(j + 8 * hi) * D_ + n] += c[j] + bias;
        }
        __syncthreads();

        norm16x128(s_x, s_x2, n2a + L * D_, n2b + L * D_, s_red, tid);

        // h = relu(x2 @ w1^T + b1) -> s_h  (16x512x128; 8 waves x 4 N-tiles)
        for (int nt = wv; nt < 32; nt += 8) {
            v8f c = gemm_tile(s_x2, D_, w1L, D_, nt * 16, D_, lane);
            int n = nt * 16 + (lane & 15);
            float bias = b1L[n];
            #pragma unroll
            for (int j = 0; j < 8; ++j)
                s_h[(j + 8 * hi) * DFF_ + n] = fmaxf(c[j] + bias, 0.f);
        }
        __syncthreads();

        // x += h @ w2^T + b2               (16x128x512)
        {
            v8f c = gemm_tile(s_h, DFF_, w2L, DFF_, wv * 16, DFF_, lane);
            int n = wv * 16 + (lane & 15);
            float bias = b2L[n];
            #pragma unroll
            for (int j = 0; j < 8; ++j) s_x[(j + 8 * hi) * D_ + n] += c[j] + bias;
        }
        __syncthreads();
    }

    // ---- final norm + linear head ----
    norm16x128(s_x, s_x2, fna, fnb, s_red, tid);
    if (tid < 16) {
        const float* row = s_x2 + tid * D_;
        float acc = out_b[0];
        for (int d = 0; d < D_; ++d) acc += row[d] * out_w[d];
        out[r0 + tid] = acc;
    }
}

extern "C" void kernel_launch(void* const* d_in, const int* in_sizes, int n_in,
                              void* d_out, int out_size, void* d_ws, size_t ws_size,
                              hipStream_t stream) {
    (void)in_sizes; (void)n_in; (void)d_ws; (void)ws_size; (void)out_size;
    const float* src  = (const float*)d_in[0];
    const int*   t    = (const int*)  d_in[1];
    const float* fc_w = (const float*)d_in[2];
    const float* fc_b = (const float*)d_in[3];
    const float* gc_w = (const float*)d_in[4];
    const float* gc_b = (const float*)d_in[5];
    const float* ln_w = (const float*)d_in[6];
    const float* ln_b = (const float*)d_in[7];
    // d_in[8..11] = qw, qb, kw, kb : mathematically unused (seq len == 1)
    const float* vw   = (const float*)d_in[12];
    const float* vb   = (const float*)d_in[13];
    const float* ow   = (const float*)d_in[14];
    const float* ob   = (const float*)d_in[15];
    const float* n1a  = (const float*)d_in[16];
    const float* n1b  = (const float*)d_in[17];
    const float* n2a  = (const float*)d_in[18];
    const float* n2b  = (const float*)d_in[19];
    const float* w1   = (const float*)d_in[20];
    const float* b1   = (const float*)d_in[21];
    const float* w2   = (const float*)d_in[22];
    const float* b2   = (const float*)d_in[23];
    const float* fna  = (const float*)d_in[24];
    const float* fnb  = (const float*)d_in[25];
    const float* outw = (const float*)d_in[26];
    const float* outb = (const float*)d_in[27];
    float* out = (float*)d_out;

    transformer_fused<<<16, 256, 0, stream>>>(
        src, t, fc_w, fc_b, gc_w, gc_b, ln_w, ln_b,
        vw, vb, ow, ob, n1a, n1b, n2a, n2b,
        w1, b1, w2, b2, fna, fnb, outw, outb, out);
}